// PathFindingNetwork_52424370815010
// MI455X (gfx1250) — compile-verified
//
#include <hip/hip_runtime.h>
#include <hip/hip_bf16.h>

typedef __attribute__((ext_vector_type(2))) float v2f;
typedef __attribute__((ext_vector_type(8))) float v8f;

#define SELF_LOOP_W 2.0f

// ---------------------------------------------------------------------------
// k0: zero the scatter accumulator, seed degree with the self-loop weight.
// ---------------------------------------------------------------------------
__global__ void gcn_init(float* __restrict__ deg, float* __restrict__ agg,
                         int n_nodes) {
  int i = blockIdx.x * blockDim.x + threadIdx.x;
  if (i < n_nodes * 32) agg[i] = 0.0f;
  if (i < n_nodes) deg[i] = SELF_LOOP_W;
}

// ---------------------------------------------------------------------------
// k1: degree accumulation over edge destinations (unit edge weights).
// ---------------------------------------------------------------------------
__global__ void gcn_degree(const long long* __restrict__ ei,
                           float* __restrict__ deg, int n_edges) {
  int i = blockIdx.x * blockDim.x + threadIdx.x;
  if (i < n_edges) {
    int d = (int)ei[(size_t)n_edges + i];
    atomicAdd(&deg[d], 1.0f);
  }
}

// ---------------------------------------------------------------------------
// k2: deg -> d^{-1/2}, in place. deg >= 2 always, no zero-guard needed.
// ---------------------------------------------------------------------------
__global__ void gcn_rsqrt(float* __restrict__ deg, int n_nodes) {
  int i = blockIdx.x * blockDim.x + threadIdx.x;
  if (i < n_nodes) deg[i] = rsqrtf(deg[i]);
}

// ---------------------------------------------------------------------------
// k3: xw = x @ W via V_WMMA_F32_16X16X4_F32, FULL 16-row tiles only.
// One wave per tile; straight-line (no guards) so EXEC is all-ones for the
// whole kernel and the store path has no branches.
// K=16 -> 4 chained K=4 WMMA steps; OUT_CH=32 -> two 16-wide N tiles.
//
// A layout (16x4 f32): lanes 0-15 -> M=lane, K={kb,kb+1}; lanes 16-31 ->
// M=lane-16, K={kb+2,kb+3}. B layout mirrored (K rows striped across lanes).
// C layout: VGPR v, M = v + 8*(lane>=16), N = lane&15.
// ---------------------------------------------------------------------------
__global__ void __launch_bounds__(32)
gcn_xw_wmma(const float* __restrict__ x, const float* __restrict__ W,
            float* __restrict__ xw) {
  const int lane  = threadIdx.x;        // 0..31, wave32
  const int m0    = blockIdx.x * 16;
  const int nlo   = lane & 15;          // N within tile / M within half
  const int khalf = (lane >> 4) * 2;    // 0 for lanes 0-15, 2 for lanes 16-31
  const int arow  = m0 + nlo;

  v8f c0 = {};
  v8f c1 = {};
#pragma unroll
  for (int kk = 0; kk < 4; ++kk) {
    const int kb = kk * 4 + khalf;      // K base for this lane-half
    v2f a, b0, b1;
    a.x  = x[arow * 16 + kb];
    a.y  = x[arow * 16 + kb + 1];
    b0.x = W[kb * 32 + nlo];            // N-tile 0 (cols 0..15)
    b0.y = W[(kb + 1) * 32 + nlo];
    b1.x = W[kb * 32 + 16 + nlo];       // N-tile 1 (cols 16..31)
    b1.y = W[(kb + 1) * 32 + 16 + nlo];
    // 8 args: (neg_a, A, neg_b, B, c_mod, C, reuse_a, reuse_b)
    c0 = __builtin_amdgcn_wmma_f32_16x16x4_f32(false, a, false, b0,
                                               (short)0, c0, false, false);
    c1 = __builtin_amdgcn_wmma_f32_16x16x4_f32(false, a, false, b1,
                                               (short)0, c1, false, false);
  }

  const int mbase = m0 + (lane >> 4) * 8;
#pragma unroll
  for (int v = 0; v < 8; ++v) {
    const int row = mbase + v;
    xw[row * 32 + nlo]      = c0[v];
    xw[row * 32 + 16 + nlo] = c1[v];
  }
}

// ---------------------------------------------------------------------------
// k3b: scalar tail for remainder rows (n_nodes % 16). Launched only if
// needed (never for N=100000). One thread per (row, out-channel).
// ---------------------------------------------------------------------------
__global__ void gcn_xw_tail(const float* __restrict__ x,
                            const float* __restrict__ W,
                            float* __restrict__ xw, int row0, int n_nodes) {
  int i = blockIdx.x * blockDim.x + threadIdx.x;
  int row = row0 + (i >> 5);
  int ch  = i & 31;
  if (row < n_nodes) {
    float acc = 0.0f;
#pragma unroll
    for (int k = 0; k < 16; ++k) acc += x[row * 16 + k] * W[k * 32 + ch];
    xw[row * 32 + ch] = acc;
  }
}

// ---------------------------------------------------------------------------
// k4: edge scatter. One wave handles edges; lane = output channel, so the
// 32-float gather of xw[src] and the 32 atomic adds into agg[dst] are each a
// single coalesced 128B burst per edge. Edge indices are loaded 32-at-a-time
// (one per lane) then broadcast with __shfl to amortize the int64 loads.
// xw/agg (12.8 MB each) are L2-resident on MI455X (192 MB L2).
// ---------------------------------------------------------------------------
__global__ void __launch_bounds__(256)
gcn_scatter(const long long* __restrict__ ei, const float* __restrict__ xw,
            const float* __restrict__ dinv, float* __restrict__ agg,
            int n_edges) {
  const int lane   = threadIdx.x & 31;
  const int wave   = (int)((blockIdx.x * blockDim.x + threadIdx.x) >> 5);
  const int nwaves = (int)((gridDim.x * blockDim.x) >> 5);

  for (int base = wave * 32; base < n_edges; base += nwaves * 32) {
    // Prefetch the next batch of this wave's index stream (global_prefetch_b8).
    __builtin_prefetch(&ei[(size_t)base + (size_t)nwaves * 32 + lane], 0, 1);

    const int e = base + lane;
    int s = 0, d = 0;
    float nrm = 0.0f;
    if (e < n_edges) {
      s   = (int)ei[e];                       // src row of edge_index
      d   = (int)ei[(size_t)n_edges + e];     // dst row of edge_index
      nrm = dinv[s] * dinv[d];
    }
    const int cnt = (n_edges - base < 32) ? (n_edges - base) : 32;
    for (int i = 0; i < cnt; ++i) {
      const int   si = __shfl(s, i, 32);
      const int   di = __shfl(d, i, 32);
      const float ni = __shfl(nrm, i, 32);
      const float val = xw[si * 32 + lane] * ni;
      atomicAdd(&agg[di * 32 + lane], val);
    }
  }
}

// ---------------------------------------------------------------------------
// k5: out = agg + 2 * dinv^2 * xw + b
// ---------------------------------------------------------------------------
__global__ void gcn_finalize(const float* __restrict__ agg,
                             const float* __restrict__ xw,
                             const float* __restrict__ dinv,
                             const float* __restrict__ b,
                             float* __restrict__ out, int n_nodes) {
  int i = blockIdx.x * blockDim.x + threadIdx.x;
  if (i < n_nodes * 32) {
    const int node = i >> 5;
    const int ch   = i & 31;
    const float di = dinv[node];
    out[i] = agg[i] + SELF_LOOP_W * di * di * xw[i] + b[ch];
  }
}

// ---------------------------------------------------------------------------
// Host-side launch. Inputs (setup_inputs order):
//   d_in[0] x          float32 [N, 16]
//   d_in[1] edge_index int64   [2, E]
//   d_in[2] W          float32 [16, 32]
//   d_in[3] b          float32 [32]
// Output: float32 [N, 32].
// Workspace layout: xw[N*32] | deg/dinv[N] | agg[N*32]  (~26 MB)
// ---------------------------------------------------------------------------
extern "C" void kernel_launch(void* const* d_in, const int* in_sizes, int n_in,
                              void* d_out, int out_size, void* d_ws,
                              size_t ws_size, hipStream_t stream) {
  const float*     x  = (const float*)d_in[0];
  const long long* ei = (const long long*)d_in[1];
  const float*     W  = (const float*)d_in[2];
  const float*     b  = (const float*)d_in[3];
  float*           out = (float*)d_out;

  const int n_nodes = in_sizes[0] / 16;
  const int n_edges = in_sizes[1] / 2;

  float* xw  = (float*)d_ws;
  float* deg = xw + (size_t)n_nodes * 32;   // becomes dinv in place
  float* agg = deg + (size_t)n_nodes;

  const int B = 256;
  const int g_nc = (n_nodes * 32 + B - 1) / B;   // node*channel grid
  const int g_e  = (n_edges + B - 1) / B;
  const int g_n  = (n_nodes + B - 1) / B;

  const int full_tiles = n_nodes / 16;           // 6250 for N=100000
  const int rem_rows   = n_nodes - full_tiles * 16;

  gcn_init<<<g_nc, B, 0, stream>>>(deg, agg, n_nodes);
  gcn_degree<<<g_e, B, 0, stream>>>(ei, deg, n_edges);
  gcn_rsqrt<<<g_n, B, 0, stream>>>(deg, n_nodes);
  if (full_tiles > 0)
    gcn_xw_wmma<<<full_tiles, 32, 0, stream>>>(x, W, xw);
  if (rem_rows > 0)
    gcn_xw_tail<<<(rem_rows * 32 + B - 1) / B, B, 0, stream>>>(
        x, W, xw, full_tiles * 16, n_nodes);
  gcn_scatter<<<2048, B, 0, stream>>>(ei, xw, deg, agg, n_edges);
  gcn_finalize<<<g_nc, B, 0, stream>>>(agg, xw, deg, b, out, n_nodes);
}